// RPRNetwork_85091892069063
// MI455X (gfx1250) — compile-verified
//
#include <hip/hip_runtime.h>

typedef float v2f __attribute__((ext_vector_type(2)));
typedef float v4f __attribute__((ext_vector_type(4)));
typedef float v8f __attribute__((ext_vector_type(8)));

#define NN    2048
#define BB    64
#define TT    128
#define CW    32          // owned columns per workgroup
#define NT    4           // 16-col tiles covering [c0-16, c0+48)
#define NL    3           // weighted (WMMA) layers; layer 3 is identity
#define SB_NC 76          // spike-buffer col stride (16B-aligned rows), origin c0-20

// Workgroup barrier that only drains DScnt (LDS consistency), NOT LOADcnt.
// __syncthreads() conservatively emits s_wait_loadcnt_dscnt 0, which would
// force our prefetched x_{t+1} global loads to complete at the first
// inter-layer barrier and destroy the software pipeline. The CDNA5 split
// counters let LDS visibility (dscnt) and global-load completion (loadcnt)
// be ordered independently; loadcnt waits are inserted by the compiler at
// the actual register uses of the prefetched data.
static __device__ __forceinline__ void wg_barrier_lds() {
    asm volatile("s_wait_dscnt 0\n\t"
                 "s_barrier_signal -1\n\t"
                 "s_barrier_wait -1" ::: "memory");
}

__global__ __launch_bounds__(128)
void snn_rpr_kernel(const float* __restrict__ x,    // [B,T,N]
                    const float* __restrict__ Wt,   // [3,N,N]
                    float* __restrict__ out)        // outs|gts|ps, each [B,T,N]
{
    __shared__ float sbuf[2][16 * SB_NC];                 // spike ping-pong, row-major [b][col]
    __shared__ float vstate[NL][NT][32][8];               // fragment order [lane][reg]
    __shared__ float trstate[NL][NT][32][8];

    const int tid  = threadIdx.x;
    const int lane = tid & 31;
    const int wv   = tid >> 5;                 // wave 0..3 owns tile wv
    const int c0   = blockIdx.x * CW;
    const int b0   = blockIdx.y * 16;
    const int SB0  = c0 - 20;                  // spike buffer col origin (multiple of 4)
    const int tc0  = c0 - 16 + 16 * wv;        // this wave's tile col start
    const bool edge = (c0 - 20 < 0) || (c0 + 52 > NN);   // uniform per-WG

    // ---- zero-init LDS (edge pads must stay finite zeros forever) ----
    for (int i = tid; i < 2 * 16 * SB_NC; i += 128) (&sbuf[0][0])[i] = 0.f;
    {
        float* vs = &vstate[0][0][0][0];
        float* ts = &trstate[0][0][0][0];
        for (int i = tid; i < NL * NT * 32 * 8; i += 128) { vs[i] = 0.f; ts[i] = 0.f; }
    }

    // ---- preload weight B-fragments (banded columns) into registers ----
    // B layout (4x16 f32): lane -> N = lane%16 ; K = 2*(lane/16) + reg
    v2f wfrag[NL][5];
    {
        const int ncol = tc0 + (lane & 15);
        #pragma unroll
        for (int li = 0; li < NL; ++li) {
            const float* Wl = Wt + (size_t)li * NN * NN;
            #pragma unroll
            for (int s = 0; s < 5; ++s) {
                const int kb = tc0 - 2 + 4 * s + 2 * (lane >> 4);
                v2f f;
                #pragma unroll
                for (int r = 0; r < 2; ++r) {
                    const int k = kb + r;
                    f[r] = (k >= 0 && k < NN && ncol >= 0 && ncol < NN)
                               ? Wl[(size_t)k * NN + ncol] : 0.f;
                }
                wfrag[li][s] = f;
            }
        }
    }

    // ---- x staging: 72 aligned cols [c0-20, c0+52), 18 float4 per batch row ----
    // 288 float4 over 128 threads: q=0..2, idx = tid + 128*q
    v4f xv[3];
    auto load_x = [&](int t) {
        #pragma unroll
        for (int q = 0; q < 3; ++q) {
            const int idx = tid + 128 * q;
            v4f val = {0.f, 0.f, 0.f, 0.f};
            if (idx < 288) {
                const int row = idx / 18, cg = idx % 18;
                const int gcol = c0 - 20 + cg * 4;
                const float* px = x + ((size_t)(b0 + row) * TT + t) * NN + gcol;
                if (!edge) {
                    val = *(const v4f*)px;               // global_load_b128
                } else {
                    #pragma unroll
                    for (int c = 0; c < 4; ++c)
                        if (gcol + c >= 0 && gcol + c < NN) val[c] = px[c];
                }
            }
            xv[q] = val;
        }
    };
    auto store_x = [&]() {
        #pragma unroll
        for (int q = 0; q < 3; ++q) {
            const int idx = tid + 128 * q;
            if (idx < 288) {
                const int row = idx / 18, cg = idx % 18;
                *(v4f*)&sbuf[0][row * SB_NC + cg * 4] = xv[q];  // ds_store_b128
            }
        }
    };

    // output-layer state in registers: idx = tid + 128*q -> (b = idx/32, col = idx%32)
    float v3[4] = {0.f, 0.f, 0.f, 0.f};
    float tr3[4] = {0.f, 0.f, 0.f, 0.f};

    const int d_col = lane & 15;          // D frag: N = lane%16
    const int d_bhi = (lane >> 4) * 8;    // D frag: M = reg + 8*(lane/16)
    const size_t BTN = (size_t)BB * TT * NN;

    load_x(0);                            // prologue of the software pipeline
    wg_barrier_lds();                     // covers LDS zero-init

    for (int t = 0; t < TT; ++t) {
        store_x();                        // publish x_t (sbuf[0] free: L2/out used sbuf[1] last)
        wg_barrier_lds();
        if (t + 1 < TT) load_x(t + 1);    // issue next step's HBM loads now; the whole
                                          // 3-layer WMMA pipeline below hides their latency

        // ---- three weighted LIF layers via banded f32 WMMA ----
        #pragma unroll
        for (int li = 0; li < NL; ++li) {
            const float* rb = sbuf[li & 1];
            float*       wb = sbuf[(li + 1) & 1];

            v8f acc = {};
            #pragma unroll
            for (int s = 0; s < 5; ++s) {
                const int kb = tc0 - 2 + 4 * s + 2 * (lane >> 4);
                // A frag (16x4 f32): M = lane%16 (batch), K = 2*(lane/16)+reg
                const v2f a = *(const v2f*)&rb[(lane & 15) * SB_NC + (kb - SB0)];
                acc = __builtin_amdgcn_wmma_f32_16x16x4_f32(
                          false, a, false, wfrag[li][s], (short)0, acc, false, false);
            }

            // LIF on the 16x16 fragment; per-tile state only touched by owner wave
            v4f* vp = (v4f*)&vstate[li][wv][lane][0];
            v4f* tp = (v4f*)&trstate[li][wv][lane][0];
            v8f v, tr;
            ((v4f*)&v)[0] = vp[0];  ((v4f*)&v)[1] = vp[1];
            ((v4f*)&tr)[0] = tp[0]; ((v4f*)&tr)[1] = tp[1];
            #pragma unroll
            for (int r = 0; r < 8; ++r) {
                const float vpre = v[r] + acc[r];
                const float sp   = (vpre >= 1.0f) ? 1.0f : 0.0f;
                v[r]  = vpre * (1.0f - sp);          // hard reset
                tr[r] = 0.9f * tr[r] + sp;           // gamma trace
                wb[(r + d_bhi) * SB_NC + (tc0 + d_col - SB0)] = sp;
            }
            vp[0] = ((v4f*)&v)[0];  vp[1] = ((v4f*)&v)[1];
            tp[0] = ((v4f*)&tr)[0]; tp[1] = ((v4f*)&tr)[1];
            wg_barrier_lds();
        }

        // ---- output layer: identity input, elementwise, coalesced stores ----
        const float* rb = sbuf[1];   // L2 wrote sbuf[1]
        #pragma unroll
        for (int q = 0; q < 4; ++q) {
            const int idx = tid + 128 * q;
            const int col = idx & (CW - 1);
            const int bb  = idx >> 5;
            const float I    = rb[bb * SB_NC + (c0 + col - SB0)];
            const float vpre = v3[q] + I;
            const float sp   = (vpre >= 1.0f) ? 1.0f : 0.0f;
            v3[q]  = vpre * (1.0f - sp);
            tr3[q] = 0.9f * tr3[q] + sp;
            const float d = vpre - 1.0f;
            const float p = expf(-50.0f * d * d);    // exp(-d^2 / (2*0.1^2))
            const size_t off = ((size_t)(b0 + bb) * TT + t) * NN + (c0 + col);
            out[off]           = sp;
            out[BTN + off]     = tr3[q];
            out[2 * BTN + off] = p;
        }
        wg_barrier_lds();
    }
}

extern "C" void kernel_launch(void* const* d_in, const int* in_sizes, int n_in,
                              void* d_out, int out_size, void* d_ws, size_t ws_size,
                              hipStream_t stream) {
    (void)in_sizes; (void)n_in; (void)out_size; (void)d_ws; (void)ws_size;
    const float* x  = (const float*)d_in[0];   // [64,128,2048] f32
    const float* Wt = (const float*)d_in[1];   // [3,2048,2048] f32
    float* out = (float*)d_out;                // 3 x [64,128,2048] f32, concatenated
    dim3 grid(NN / CW, BB / 16);               // 64 x 4 = 256 workgroups
    snn_rpr_kernel<<<grid, 128, 0, stream>>>(x, Wt, out);
}